// NegNCE_39994735460887
// MI455X (gfx1250) — compile-verified
//
#include <hip/hip_runtime.h>
#include <stdint.h>

// ---------------- problem constants ----------------
#define B_ROWS   1024
#define VCOLS    100000
#define KNEG     100
#define NPOS     (KNEG + 1)         // 101
#define EPSF     1e-6f

// ---------------- kernel tuning --------------------
#define TILE     2048               // floats per TDM tile (8 KB)
#define NBINS    8192               // 13-bit histogram
#define NCHUNK   256                // chunks for WMMA suffix scan (32 bins/chunk)
#define BINS_PER_CHUNK (NBINS / NCHUNK)
#define CAND_CAP 1024               // threshold-bin candidate buffer
#define BLK      256                // threads in top-k block (8 waves)

typedef unsigned int u32;
typedef unsigned long long u64;
typedef __attribute__((ext_vector_type(2))) float v2f;
typedef __attribute__((ext_vector_type(8))) float v8f;
typedef __attribute__((ext_vector_type(4))) u32  v4u;
typedef __attribute__((ext_vector_type(8))) int  v8i;
typedef __attribute__((ext_vector_type(4))) int  v4i;

// ---------------- helpers --------------------------
__device__ __forceinline__ u32 hash_u32(u32 x) {
    x ^= x >> 16; x *= 0x7feb352du;
    x ^= x >> 15; x *= 0x846ca68bu;
    x ^= x >> 16; return x;
}

// Gumbel(0,1) noise from a counter-based hash (statistical stand-in for
// jax.random.gumbel; bitwise reproduction of threefry is not required).
__device__ __forceinline__ float gumbel_noise(u32 r, u32 v) {
    u32 h = hash_u32(v ^ (r * 0x9E3779B9u) ^ 0xA5A5A5A5u);
    float un = (float)((h >> 8) + 1u) * (1.0f / 16777218.0f); // (0,1)
    return -__logf(-__logf(un));
}

// float -> order-preserving uint32
__device__ __forceinline__ u32 sortable_key(float f) {
    u32 b = __float_as_uint(f);
    return (b & 0x80000000u) ? ~b : (b | 0x80000000u);
}

// low 32 bits of a generic pointer to LDS == byte offset in LDS
__device__ __forceinline__ u32 lds_off(const void* p) {
    return (u32)(uintptr_t)p;
}

// Issue one TDM 1-D tile copy: len f32 elements from global `src` into LDS.
// D# group0: count=1, lds_addr, 57-bit global addr, type=2 ("image").
// D# group1: data_size=4B, tensor_dim0=tile_dim0=len, tensor_dim1=1.
// amdgpu-toolchain (clang-23) 6-arg builtin form:
//   (v4u g0, v8i g1, v4i g2, v4i g3, v8i g4, i32 cpol)
__device__ __forceinline__ void tdm_load_tile(const float* src, u32 lds_byte, int len) {
    u64 ga = (u64)(uintptr_t)src;
    v4u g0;
    g0[0] = 1u;                                  // count=1, user descriptor
    g0[1] = lds_byte;                            // lds_addr
    g0[2] = (u32)ga;                             // global_addr[31:0]
    g0[3] = (u32)(ga >> 32) | (2u << 30);        // global_addr[56:32] | type=2
    v8i g1;
    g1[0] = (int)(2u << 16);                     // data_size = 2 (4 bytes)
    g1[1] = (int)(((u32)len & 0xFFFFu) << 16);   // tensor_dim0[15:0] @ bits 63:48
    g1[2] = (int)(1u << 16);                     // tensor_dim0 hi=0, tensor_dim1=1
    g1[3] = (int)(((u32)len) << 16);             // tile_dim0 @ bits 127:112
    g1[4] = 0;                                   // tile_dim1=0 (unused), tile_dim2=0
    g1[5] = (int)(u32)len;                       // tensor_dim0_stride lo32
    g1[6] = 0;
    g1[7] = 0;
    v4i z4 = {0, 0, 0, 0};
    v8i z8 = {0, 0, 0, 0, 0, 0, 0, 0};
    __builtin_amdgcn_tensor_load_to_lds(g0, g1, z4, z4, z8, 0);
}

// ============================================================
// Kernel 1: per-row Gumbel top-K via 13-bit histogram select.
// One 256-thread block per row. Data streamed via TDM double buffer.
// ============================================================
__global__ __launch_bounds__(BLK)
void nce_topk_kernel(const float* __restrict__ noise,
                     const int* __restrict__ target_id,
                     int* __restrict__ items /* [B][101] */) {
    __shared__ float tile[2][TILE];
    __shared__ u32   hist[NBINS];
    __shared__ float chunksum[NCHUNK];
    __shared__ float suffc[NCHUNK];
    __shared__ float m2buf[NCHUNK];
    __shared__ u64   cand[CAND_CAP];
    __shared__ int   sh_bstar, sh_D, sh_direct, sh_candcnt;

    const int r      = blockIdx.x;
    const int tid    = (int)threadIdx.x;
    const int target = target_id[r];
    const float* row = noise + (size_t)r * VCOLS;
    const int ntiles = (VCOLS + TILE - 1) / TILE;
    // force a *scalar* branch for the DMA-issuing wave (wave 0): TDM issue
    // ignores EXEC, so a merely EXEC-masked path would issue 8 duplicate DMAs.
    const int wave_first_tid = __builtin_amdgcn_readfirstlane(tid);

    for (int pass = 0; pass < 2; ++pass) {
        // ---- per-pass init ----
        if (pass == 0) {
            for (int i = tid; i < NBINS; i += BLK) hist[i] = 0u;
        } else {
            for (int i = tid; i < CAND_CAP; i += BLK) cand[i] = 0ull;
            if (tid == 0) { sh_direct = 0; sh_candcnt = 0; }
        }
        __syncthreads();
        const int bstar_l = (pass == 1) ? sh_bstar : 0;

        // ---- stream row through LDS via TDM, double buffered ----
        if (wave_first_tid == 0) {
            tdm_load_tile(row, lds_off(&tile[0][0]), min(TILE, VCOLS));
        }
        for (int t = 0; t < ntiles; ++t) {
            const int nt = t + 1;
            if (wave_first_tid == 0) {
                if (nt < ntiles) {
                    const int pos = nt * TILE;
                    tdm_load_tile(row + pos, lds_off(&tile[nt & 1][0]),
                                  min(TILE, VCOLS - pos));
                    __builtin_amdgcn_s_wait_tensorcnt(1); // tile t done (in-order)
                } else {
                    __builtin_amdgcn_s_wait_tensorcnt(0);
                }
            }
            __syncthreads();              // publish LDS tile to all waves

            const float* tb = tile[t & 1];
            const int base  = t * TILE;
            const int len   = min(TILE, VCOLS - base);
            for (int i = tid; i < len; i += BLK) {
                const int v = base + i;
                if (v == target) continue;               // mask target (-inf)
                const float key = tb[i] + gumbel_noise((u32)r, (u32)v);
                const u32 sk = sortable_key(key);
                const int bin = (int)(sk >> 19);         // top 13 bits
                if (pass == 0) {
                    atomicAdd(&hist[bin], 1u);
                } else {
                    if (bin > bstar_l) {
                        const int p = atomicAdd(&sh_direct, 1);
                        items[r * NPOS + 1 + p] = v;     // definitely in top-K
                    } else if (bin == bstar_l) {
                        const int c = atomicAdd(&sh_candcnt, 1);
                        if (c < CAND_CAP)
                            cand[c] = ((u64)sk << 32) | (u32)v;
                    }
                }
            }
            __syncthreads();              // tile consumed before re-fill
        }

        if (pass == 0) {
            // ---- chunk sums (exact in f32: counts <= V < 2^24) ----
            float s = 0.0f;
            for (int i = 0; i < BINS_PER_CHUNK; ++i)
                s += (float)hist[tid * BINS_PER_CHUNK + i];
            chunksum[tid] = s;
            __syncthreads();

            // ---- strict suffix sum of 256 chunk counts on the matrix core:
            //      S = H*SL + (SLr*H)*J ; three 16x16x16 f32 products,
            //      each as 4 chained V_WMMA_F32_16X16X4_F32. wave 0 only
            //      (EXEC all ones within the wave, as WMMA requires). ----
            if (tid < 32) {
                const int l    = tid;
                const int Mrow = l & 15;           // A-operand row
                const int Ncol = l & 15;           // B/C/D column
                const int kb   = (l >> 4) << 1;    // K sub-lane base: 0 or 2
                v8f acc  = {};
                v8f macc = {};
                #pragma unroll
                for (int kk = 0; kk < 4; ++kk) {   // acc = H x SL
                    const int k0 = kk * 4 + kb, k1 = k0 + 1;
                    v2f a, b;
                    a[0] = chunksum[Mrow * 16 + k0];
                    a[1] = chunksum[Mrow * 16 + k1];
                    b[0] = (k0 > Ncol) ? 1.0f : 0.0f;
                    b[1] = (k1 > Ncol) ? 1.0f : 0.0f;
                    acc = __builtin_amdgcn_wmma_f32_16x16x4_f32(
                        false, a, false, b, (short)0, acc, false, false);
                }
                #pragma unroll
                for (int kk = 0; kk < 4; ++kk) {   // macc = SLr x H
                    const int k0 = kk * 4 + kb, k1 = k0 + 1;
                    v2f a, b;
                    a[0] = (k0 > Mrow) ? 1.0f : 0.0f;
                    a[1] = (k1 > Mrow) ? 1.0f : 0.0f;
                    b[0] = chunksum[k0 * 16 + Ncol];
                    b[1] = chunksum[k1 * 16 + Ncol];
                    macc = __builtin_amdgcn_wmma_f32_16x16x4_f32(
                        false, a, false, b, (short)0, macc, false, false);
                }
                // spill macc (C/D layout: VGPR g -> row g + 8*(lane/16))
                #pragma unroll
                for (int g = 0; g < 8; ++g)
                    m2buf[(g + ((l >> 4) << 3)) * 16 + Ncol] = macc[g];
                #pragma unroll
                for (int kk = 0; kk < 4; ++kk) {   // acc += (SLr*H) x J
                    const int k0 = kk * 4 + kb, k1 = k0 + 1;
                    v2f a, b;
                    a[0] = m2buf[Mrow * 16 + k0];
                    a[1] = m2buf[Mrow * 16 + k1];
                    b[0] = 1.0f; b[1] = 1.0f;
                    acc = __builtin_amdgcn_wmma_f32_16x16x4_f32(
                        false, a, false, b, (short)0, acc, false, false);
                }
                #pragma unroll
                for (int g = 0; g < 8; ++g)
                    suffc[(g + ((l >> 4) << 3)) * 16 + Ncol] = acc[g];
            }
            __syncthreads();

            // ---- locate K-th-largest bin ----
            if (tid == 0) {
                int cstar = 0;
                for (int c = NCHUNK - 1; c >= 0; --c) {
                    const float ab = suffc[c];
                    if (ab < (float)KNEG && ab + chunksum[c] >= (float)KNEG) {
                        cstar = c; break;
                    }
                }
                int above = (int)suffc[cstar];
                int bstar = cstar * BINS_PER_CHUNK;
                for (int b = BINS_PER_CHUNK - 1; b >= 0; --b) {
                    const int h = (int)hist[cstar * BINS_PER_CHUNK + b];
                    if (above + h >= KNEG) { bstar = cstar * BINS_PER_CHUNK + b; break; }
                    above += h;
                }
                sh_bstar = bstar;   // threshold bin
                sh_D     = above;   // count strictly above it (< K)
            }
            __syncthreads();
        } else {
            // ---- bitonic sort candidates ascending (pad=0 sinks) ----
            for (int k = 2; k <= CAND_CAP; k <<= 1) {
                for (int j = k >> 1; j > 0; j >>= 1) {
                    for (int i = tid; i < CAND_CAP; i += BLK) {
                        const int ixj = i ^ j;
                        if (ixj > i) {
                            const bool up = ((i & k) == 0);
                            const u64 a = cand[i], b = cand[ixj];
                            if ((a > b) == up) { cand[i] = b; cand[ixj] = a; }
                        }
                    }
                    __syncthreads();
                }
            }
            const int need = KNEG - sh_D;
            for (int j = tid; j < need; j += BLK)
                items[r * NPOS + 1 + sh_D + j] =
                    (int)(cand[CAND_CAP - 1 - j] & 0xFFFFFFFFu);
            if (tid == 0) items[r * NPOS] = target;      // slot 0 = positive
        }
    }
}

// ============================================================
// Kernel 2: per-row scoring: gather 101 logits, double softmax,
// NCE likelihood, per-row sum of log-likelihood.
// ============================================================
__global__ __launch_bounds__(128)
void nce_score_kernel(const float* __restrict__ noise,
                      const float* __restrict__ actual,
                      const int* __restrict__ items,
                      float* __restrict__ rowloss) {
    __shared__ float red[128];
    __shared__ float nl[NPOS];
    __shared__ float al[NPOS];
    const int r = blockIdx.x, tid = (int)threadIdx.x;

    if (tid < NPOS) {
        int it = items[r * NPOS + tid];
        it = min(max(it, 0), VCOLS - 1);
        nl[tid] = noise[(size_t)r * VCOLS + it];
        al[tid] = actual[r * NPOS + tid];
    }
    __syncthreads();

    // softmax(noise picks)
    red[tid] = (tid < NPOS) ? nl[tid] : -3.4e38f;
    __syncthreads();
    for (int s = 64; s > 0; s >>= 1) {
        if (tid < s) red[tid] = fmaxf(red[tid], red[tid + s]);
        __syncthreads();
    }
    const float mnoise = red[0]; __syncthreads();
    const float e1 = (tid < NPOS) ? __expf(nl[tid] - mnoise) : 0.0f;
    red[tid] = e1; __syncthreads();
    for (int s = 64; s > 0; s >>= 1) {
        if (tid < s) red[tid] += red[tid + s];
        __syncthreads();
    }
    const float snoise = red[0]; __syncthreads();

    // softmax(actual)
    red[tid] = (tid < NPOS) ? al[tid] : -3.4e38f;
    __syncthreads();
    for (int s = 64; s > 0; s >>= 1) {
        if (tid < s) red[tid] = fmaxf(red[tid], red[tid + s]);
        __syncthreads();
    }
    const float mact = red[0]; __syncthreads();
    const float e2 = (tid < NPOS) ? __expf(al[tid] - mact) : 0.0f;
    red[tid] = e2; __syncthreads();
    for (int s = 64; s > 0; s >>= 1) {
        if (tid < s) red[tid] += red[tid + s];
        __syncthreads();
    }
    const float sact = red[0]; __syncthreads();

    float part = 0.0f;
    if (tid < NPOS) {
        const float np_ = e1 / snoise;
        const float ap_ = e2 / sact;
        const float deno = (float)KNEG * np_ + ap_ + EPSF;
        float lik = ((tid == 0) ? ap_ : np_) / deno;
        if (lik == 1.0f) lik = 1.0f + EPSF;
        part = __logf(lik);
    }
    red[tid] = part; __syncthreads();
    for (int s = 64; s > 0; s >>= 1) {
        if (tid < s) red[tid] += red[tid + s];
        __syncthreads();
    }
    if (tid == 0) rowloss[r] = red[0];
}

// ============================================================
// Kernel 3: deterministic final reduction -> scalar loss.
// ============================================================
__global__ __launch_bounds__(256)
void nce_finalize_kernel(const float* __restrict__ rowloss,
                         float* __restrict__ out) {
    __shared__ float red[256];
    const int tid = (int)threadIdx.x;
    float s = 0.0f;
    for (int i = tid; i < B_ROWS; i += 256) s += rowloss[i];
    red[tid] = s; __syncthreads();
    for (int st = 128; st > 0; st >>= 1) {
        if (tid < st) red[tid] += red[tid + st];
        __syncthreads();
    }
    if (tid == 0) out[0] = -red[0] / (float)(B_ROWS * NPOS);
}

// ============================================================
extern "C" void kernel_launch(void* const* d_in, const int* in_sizes, int n_in,
                              void* d_out, int out_size, void* d_ws, size_t ws_size,
                              hipStream_t stream) {
    (void)in_sizes; (void)n_in; (void)out_size; (void)ws_size;
    const float* noise  = (const float*)d_in[0];   // [1024, 100000] f32
    const float* actual = (const float*)d_in[1];   // [1024, 101] f32
    const int*   tgt    = (const int*)d_in[2];     // [1024] int

    int*   items   = (int*)d_ws;                   // [1024][101] int
    size_t ofs     = ((size_t)B_ROWS * NPOS * sizeof(int) + 255u) & ~(size_t)255u;
    float* rowloss = (float*)((char*)d_ws + ofs);  // [1024] f32
    float* out     = (float*)d_out;

    nce_topk_kernel<<<B_ROWS, BLK, 0, stream>>>(noise, tgt, items);
    nce_score_kernel<<<B_ROWS, 128, 0, stream>>>(noise, actual, items, rowloss);
    nce_finalize_kernel<<<1, 256, 0, stream>>>(rowloss, d_out ? out : rowloss);
}